// Adaptive_Grid_46978352283656
// MI455X (gfx1250) — compile-verified
//
#include <hip/hip_runtime.h>
#include <hip/hip_bf16.h>

#define VERTEX_NUM 8192
#define GRID_RES   64
#define NBINS      (GRID_RES * GRID_RES * GRID_RES)   // 262144
#define SCAN_THREADS 1024
#define BINS_PER_THREAD (NBINS / SCAN_THREADS)        // 256

typedef __attribute__((ext_vector_type(2))) float v2f;
typedef __attribute__((ext_vector_type(8))) float v8f;

// ---- ordered-uint encoding so float min/max can use integer atomics ----
__device__ __forceinline__ unsigned fenc(float f) {
    unsigned u = __float_as_uint(f);
    return (u & 0x80000000u) ? ~u : (u | 0x80000000u);
}
__device__ __forceinline__ float fdec(unsigned u) {
    return __uint_as_float((u & 0x80000000u) ? (u ^ 0x80000000u) : ~u);
}

// ---------------- pass 0: init hist + minmax ----------------
__global__ void k_init(unsigned* __restrict__ minmax, int* __restrict__ hist) {
    int i = blockIdx.x * blockDim.x + threadIdx.x;
    if (i < NBINS) hist[i] = 0;
    if (i < 3)       minmax[i] = 0xFFFFFFFFu;  // encoded mins -> +inf
    else if (i < 6)  minmax[i] = 0u;           // encoded maxs -> -inf
}

// ---------------- pass 1: per-dim min/max ----------------
__global__ void k_minmax(const float* __restrict__ xyz, int npts,
                         unsigned* __restrict__ minmax) {
    int tid    = blockIdx.x * blockDim.x + threadIdx.x;
    int stride = gridDim.x * blockDim.x;
    unsigned mn[3] = {0xFFFFFFFFu, 0xFFFFFFFFu, 0xFFFFFFFFu};
    unsigned mx[3] = {0u, 0u, 0u};
    for (int i = tid; i < npts; i += stride) {
        #pragma unroll
        for (int d = 0; d < 3; ++d) {
            unsigned e = fenc(xyz[3 * i + d]);
            mn[d] = min(mn[d], e);
            mx[d] = max(mx[d], e);
        }
    }
    #pragma unroll
    for (int s = 16; s >= 1; s >>= 1) {
        #pragma unroll
        for (int d = 0; d < 3; ++d) {
            mn[d] = min(mn[d], (unsigned)__shfl_xor((int)mn[d], s, 32));
            mx[d] = max(mx[d], (unsigned)__shfl_xor((int)mx[d], s, 32));
        }
    }
    if ((threadIdx.x & 31) == 0) {
        #pragma unroll
        for (int d = 0; d < 3; ++d) {
            atomicMin(&minmax[d], mn[d]);
            atomicMax(&minmax[3 + d], mx[d]);
        }
    }
}

__device__ __forceinline__ int voxel_id(const float* __restrict__ xyz, int i,
                                        const unsigned* __restrict__ minmax) {
    int v[3];
    #pragma unroll
    for (int d = 0; d < 3; ++d) {
        float mn = fdec(minmax[d]);
        float mx = fdec(minmax[3 + d]);
        float ext = mx - mn;
        float inv = 1.0f / ((ext > 0.0f) ? ext : 1.0f);
        float u = (xyz[3 * i + d] - mn) * inv;
        u = fminf(fmaxf(u, 0.0f), 1.0f - 1e-6f);
        v[d] = (int)(u * (float)GRID_RES);
    }
    return (v[0] * GRID_RES + v[1]) * GRID_RES + v[2];
}

// ---------------- pass 2: voxel histogram ----------------
__global__ void k_hist(const float* __restrict__ xyz, int npts,
                       const unsigned* __restrict__ minmax, int* __restrict__ hist) {
    int i = blockIdx.x * blockDim.x + threadIdx.x;
    if (i >= npts) return;
    atomicAdd(&hist[voxel_id(xyz, i, minmax)], 1);
}

// ---------------- pass 3: exclusive scan over 262144 bins (1 block) ----------------
__global__ void __launch_bounds__(SCAN_THREADS) k_scan(int* __restrict__ hist) {
    __shared__ int sums[SCAN_THREADS];
    int t = threadIdx.x;
    int base = t * BINS_PER_THREAD;
    int s = 0;
    for (int j = 0; j < BINS_PER_THREAD; ++j) s += hist[base + j];
    sums[t] = s;
    __syncthreads();
    for (int off = 1; off < SCAN_THREADS; off <<= 1) {
        int v = (t >= off) ? sums[t - off] : 0;
        __syncthreads();
        sums[t] += v;
        __syncthreads();
    }
    int run = sums[t] - s;   // exclusive prefix of this thread's chunk
    for (int j = 0; j < BINS_PER_THREAD; ++j) {
        int c = hist[base + j];
        hist[base + j] = run;
        run += c;
    }
}

// ---------------- pass 4: scatter points into voxel order ----------------
__global__ void k_scatter(const float* __restrict__ xyz, int npts,
                          const unsigned* __restrict__ minmax,
                          int* __restrict__ hist, int* __restrict__ order) {
    int i = blockIdx.x * blockDim.x + threadIdx.x;
    if (i >= npts) return;
    int pos = atomicAdd(&hist[voxel_id(xyz, i, minmax)], 1);
    order[pos] = i;
}

// ---------------- pass 5: pick 8192 evenly-spaced vertices ----------------
__global__ void k_select(const float* __restrict__ xyz, const int* __restrict__ order,
                         int npts, float4* __restrict__ verts4,
                         float* __restrict__ vert_out) {
    int v = blockIdx.x * blockDim.x + threadIdx.x;
    if (v >= VERTEX_NUM) return;
    int sel = (int)((double)v * (double)(npts - 1) / (double)(VERTEX_NUM - 1));
    if (sel > npts - 1) sel = npts - 1;
    int idx = order[sel];
    float x = xyz[3 * idx], y = xyz[3 * idx + 1], z = xyz[3 * idx + 2];
    verts4[v] = make_float4(x, y, z, x * x + y * y + z * z);
    vert_out[3 * v]     = x;
    vert_out[3 * v + 1] = y;
    vert_out[3 * v + 2] = z;
}

// ---------------- pass 6: 1-NN via V_WMMA_F32_16X16X4_F32 ----------------
// One wave owns 16 points.  A (16x4, M x K) = [-2x, -2y, -2z, 1] per point;
// B (4x16, K x N) = [vx, vy, vz, |v|^2] per vertex; C = 0.
// => D[m,n] = |v_n|^2 - 2 * p_m . v_n  (argmin-equivalent distance).
__global__ void __launch_bounds__(256) k_knn(const float* __restrict__ xyz,
                                             const float4* __restrict__ verts4,
                                             int* __restrict__ p2v, int npts) {
    int lane   = threadIdx.x & 31;
    int wave   = (blockIdx.x * blockDim.x + threadIdx.x) >> 5;
    int ptBase = wave * 16;
    if (ptBase >= npts) return;                 // wave-uniform

    int  col = lane & 15;                       // N (B/C/D) or M (A) within tile
    bool hi  = lane >= 16;

    // A: lanes 0-15 hold K=0,1; lanes 16-31 hold K=2,3  (ISA 32-bit A 16x4 layout)
    const float* p = xyz + 3 * (ptBase + col);
    float px = p[0], py = p[1], pz = p[2];
    v2f A;
    A.x = hi ? (-2.0f * pz) : (-2.0f * px);
    A.y = hi ? 1.0f         : (-2.0f * py);

    v8f czero = {};                             // loop-invariant C = 0 (inline)

    float best[8];
    int   bidx[8];
    #pragma unroll
    for (int k = 0; k < 8; ++k) { best[k] = __builtin_inff(); bidx[k] = 0; }

    #pragma unroll 2
    for (int vt = 0; vt < VERTEX_NUM; vt += 16) {
        float4 vv = verts4[vt + col];           // broadcast across lane halves
        v2f B;                                  // B: lanes 0-15 K=0,1; 16-31 K=2,3
        B.x = hi ? vv.z : vv.x;
        B.y = hi ? vv.w : vv.y;

        // 8 args: (neg_a, A, neg_b, B, c_mod, C, reuse_a, reuse_b)
        v8f D = __builtin_amdgcn_wmma_f32_16x16x4_f32(
            false, A, false, B, (short)0, czero, false, false);

        int n = vt + col;
        #pragma unroll
        for (int k = 0; k < 8; ++k) {
            if (D[k] < best[k]) { best[k] = D[k]; bidx[k] = n; }
        }
    }

    // Cross-lane argmin within each 16-lane half (rows M=k in lanes 0-15,
    // rows M=k+8 in lanes 16-31 per the C/D VGPR layout).
    #pragma unroll
    for (int k = 0; k < 8; ++k) {
        float bv = best[k];
        int   bi = bidx[k];
        #pragma unroll
        for (int m = 8; m >= 1; m >>= 1) {
            float ov = __shfl_xor(bv, m, 32);
            int   oi = __shfl_xor(bi, m, 32);
            if (ov < bv || (ov == bv && oi < bi)) { bv = ov; bi = oi; }
        }
        if ((lane & 15) == 0) {
            int row = k + (hi ? 8 : 0);
            p2v[ptBase + row] = bi;
        }
    }
}

extern "C" void kernel_launch(void* const* d_in, const int* in_sizes, int n_in,
                              void* d_out, int out_size, void* d_ws, size_t ws_size,
                              hipStream_t stream) {
    const float* xyz = (const float*)d_in[0];
    int npts = in_sizes[0] / 3;                 // 1048576

    // workspace layout (~5.4 MB)
    char*     ws      = (char*)d_ws;
    float4*   verts4  = (float4*)ws;                                    // 128 KB
    unsigned* minmax  = (unsigned*)(ws + (size_t)VERTEX_NUM * 16);      // 6 uints
    int*      hist    = (int*)(ws + (size_t)VERTEX_NUM * 16 + 128);     // 1 MB
    int*      order   = hist + NBINS;                                   // 4 MB

    float* vert_out = (float*)d_out;                   // [8192,3] f32
    int*   p2v      = (int*)d_out + VERTEX_NUM * 3;    // [npts] i32 (raw bits)

    k_init   <<<(NBINS + 255) / 256, 256, 0, stream>>>(minmax, hist);
    k_minmax <<<1024, 256, 0, stream>>>(xyz, npts, minmax);
    k_hist   <<<(npts + 255) / 256, 256, 0, stream>>>(xyz, npts, minmax, hist);
    k_scan   <<<1, SCAN_THREADS, 0, stream>>>(hist);
    k_scatter<<<(npts + 255) / 256, 256, 0, stream>>>(xyz, npts, minmax, hist, order);
    k_select <<<(VERTEX_NUM + 255) / 256, 256, 0, stream>>>(xyz, order, npts, verts4, vert_out);

    int waves  = npts / 16;                     // one wave per 16 points
    int blocks = (waves * 32) / 256;            // 8 waves per block
    k_knn    <<<blocks, 256, 0, stream>>>(xyz, verts4, p2v, npts);
}